// uncertainty_estimator_21114059227766
// MI455X (gfx1250) — compile-verified
//
#include <hip/hip_runtime.h>
#include <hip/hip_bf16.h>

// GCN (2 conv layers + linear head) for MI455X / gfx1250.
//
// Memory-bound design: edge index streams from HBM (~640 MB total across 3
// edge passes); all node-state tables (padded features 32 MB, accumulators
// 32 MB, deg_inv_sqrt 4 MB) fit in the 192 MB L2, so random gathers and the
// f32 scatter-atomics stay on-chip. The dense 5x5 node transforms use
// V_WMMA_F32_16X16X4_F32 (one wave = 16 nodes, K split 4+1 with zero pad).

typedef __attribute__((ext_vector_type(2))) float v2f;
typedef __attribute__((ext_vector_type(8))) float v8f;

#define FDIM 5
#define PSTR 8   // padded per-node row stride (floats) -> 32B aligned rows

// ---------------------------------------------------------------- degree ----
__global__ void deg_init_kernel(float* __restrict__ deg,
                                float* __restrict__ accum, int n) {
  int i = blockIdx.x * blockDim.x + threadIdx.x;
  if (i >= n) return;
  deg[i] = 1.0f;  // self-loop contributes 1 to every node's degree
  float4 z = make_float4(0.f, 0.f, 0.f, 0.f);
  float4* p = reinterpret_cast<float4*>(accum + (size_t)i * PSTR);
  p[0] = z;
  p[1] = z;
}

__global__ void deg_edge_kernel(const int* __restrict__ dst,
                                float* __restrict__ deg, int e_cnt) {
  int e = blockIdx.x * blockDim.x + threadIdx.x;
  if (e >= e_cnt) return;
  int d = dst[e];
  __hip_atomic_fetch_add(&deg[d], 1.0f, __ATOMIC_RELAXED,
                         __HIP_MEMORY_SCOPE_AGENT);
}

__global__ void deg_fin_kernel(float* __restrict__ deg, int n) {
  int i = blockIdx.x * blockDim.x + threadIdx.x;
  if (i >= n) return;
  float d = deg[i];
  deg[i] = (d > 0.f) ? rsqrtf(d) : 0.f;  // becomes deg_inv_sqrt in place
}

// ---------------------------------------------- WMMA node transform ---------
// out[node, 0..4] = in[node, 0..4] @ Wm (FDIM x FDIM, row-major).
// One wave handles 16 nodes via V_WMMA_F32_16X16X4_F32:
//   A (16x4 f32): vgpr0 = K{0|2}, vgpr1 = K{1|3}; lanes 0-15 -> M, 16-31 -> M.
//   B (4x16 f32): vgpr0 = K{0|2}, vgpr1 = K{1|3}; N striped across lanes.
//   C/D (16x16 f32): vgpr r holds rows {r, r+8}; N = lane%16.
// K=5 handled as K=4 WMMA + K=1 WMMA (zero-padded second slice).
// All predication before the WMMAs is done with selects (not EXEC), so EXEC
// stays all-1s through the matrix ops. Safe to run in place (in == out,
// in_stride == PSTR): each wave reads only the 16 rows it later writes.
__global__ void __launch_bounds__(256)
wmma_transform_kernel(const float* __restrict__ in, int in_stride,
                      const float* __restrict__ Wm,
                      float* __restrict__ out, int n_nodes) {
  int wave = (int)((blockIdx.x * blockDim.x + threadIdx.x) >> 5);
  int lane = (int)(threadIdx.x & 31);
  int base = wave * 16;
  if (base >= n_nodes) return;  // wave-uniform: whole wave exits together

  int  m  = lane & 15;
  bool lo = lane < 16;

  // A-matrix loads: clamp node so partial tiles read in-bounds rows.
  int node = base + m;
  if (node >= n_nodes) node = n_nodes - 1;
  const float* row = in + (size_t)node * in_stride;
  int k0 = lo ? 0 : 2;
  float r0 = row[k0];
  float r1 = row[k0 + 1];
  float r4 = row[4];          // unconditional load, masked by select below

  v2f a, a2, b, b2;
  a.x  = r0;
  a.y  = r1;
  a2.x = lo ? r4 : 0.0f;      // 5th input channel = K-slice of second WMMA
  a2.y = 0.0f;

  // B-matrix: clamped loads + select masking (columns >= FDIM are zero pad).
  bool valid = (m < FDIM);
  int  mc    = valid ? m : 0;
  float w0 = Wm[(lo ? 0 : 2) * FDIM + mc];
  float w1 = Wm[(lo ? 1 : 3) * FDIM + mc];
  float w4 = Wm[4 * FDIM + mc];
  b.x  = valid ? w0 : 0.0f;
  b.y  = valid ? w1 : 0.0f;
  b2.x = (valid && lo) ? w4 : 0.0f;
  b2.y = 0.0f;

  v8f c = {};
  c = __builtin_amdgcn_wmma_f32_16x16x4_f32(false, a,  false, b,
                                            (short)0, c, false, false);
  c = __builtin_amdgcn_wmma_f32_16x16x4_f32(false, a2, false, b2,
                                            (short)0, c, false, false);

  // D tile: this lane holds out[mb + r][m] in c[r], r = 0..7.
  int    mb = base + (lo ? 0 : 8);
  float* po = out + (size_t)mb * PSTR + m;
  if (base + 16 <= n_nodes) {          // full tile (always, for N = 1M)
    if (valid) {
#pragma unroll
      for (int r = 0; r < 8; ++r) po[r * PSTR] = c[r];  // imm offsets r*32B
    }
  } else {                             // generic partial-tile fallback
#pragma unroll
    for (int r = 0; r < 8; ++r) {
      if (valid && (mb + r) < n_nodes) po[r * PSTR] = c[r];
    }
  }
}

// ------------------------------------------------------- edge aggregate -----
// accum[dst] += xw[src] * (dinv[src] * dinv[dst]);  5 f32 L2 atomics / edge.
__global__ void edge_agg_kernel(const int* __restrict__ src,
                                const int* __restrict__ dst,
                                const float* __restrict__ dinv,
                                const float* __restrict__ xw,
                                float* __restrict__ accum, int e_cnt) {
  int e = blockIdx.x * blockDim.x + threadIdx.x;
  if (e >= e_cnt) return;
  int s = src[e];
  int d = dst[e];
  float nrm = dinv[s] * dinv[d];
  const float* xr = xw + (size_t)s * PSTR;
  float4 v = *reinterpret_cast<const float4*>(xr);  // 32B-aligned row
  float  v4 = xr[4];
  float* ar = accum + (size_t)d * PSTR;
  __hip_atomic_fetch_add(ar + 0, v.x * nrm, __ATOMIC_RELAXED, __HIP_MEMORY_SCOPE_AGENT);
  __hip_atomic_fetch_add(ar + 1, v.y * nrm, __ATOMIC_RELAXED, __HIP_MEMORY_SCOPE_AGENT);
  __hip_atomic_fetch_add(ar + 2, v.z * nrm, __ATOMIC_RELAXED, __HIP_MEMORY_SCOPE_AGENT);
  __hip_atomic_fetch_add(ar + 3, v.w * nrm, __ATOMIC_RELAXED, __HIP_MEMORY_SCOPE_AGENT);
  __hip_atomic_fetch_add(ar + 4, v4  * nrm, __ATOMIC_RELAXED, __HIP_MEMORY_SCOPE_AGENT);
}

// ---------------------------------------------------------- finalizers ------
// h = relu(accum + selfloop + bias) written IN PLACE over xw (bufA);
// accum (bufB) re-zeroed for the next layer.
__global__ void finalize1_kernel(float* __restrict__ accum,
                                 float* __restrict__ xw,
                                 const float* __restrict__ dinv,
                                 const float* __restrict__ bias, int n) {
  int i = blockIdx.x * blockDim.x + threadIdx.x;
  if (i >= n) return;
  float d2 = dinv[i] * dinv[i];
  float* xr = xw + (size_t)i * PSTR;
  float* ar = accum + (size_t)i * PSTR;
#pragma unroll
  for (int f = 0; f < FDIM; ++f) {
    float h = ar[f] + xr[f] * d2 + bias[f];
    xr[f] = fmaxf(h, 0.0f);
    ar[f] = 0.0f;
  }
}

__global__ void finalize2_kernel(const float* __restrict__ accum,
                                 const float* __restrict__ xw,
                                 const float* __restrict__ dinv,
                                 const float* __restrict__ bias,
                                 const float* __restrict__ Wl,
                                 const float* __restrict__ bl,
                                 float* __restrict__ out, int n) {
  int i = blockIdx.x * blockDim.x + threadIdx.x;
  if (i >= n) return;
  float d2 = dinv[i] * dinv[i];
  const float* xr = xw + (size_t)i * PSTR;
  const float* ar = accum + (size_t)i * PSTR;
  float acc = bl[0];
#pragma unroll
  for (int f = 0; f < FDIM; ++f) {
    float h = fmaxf(ar[f] + xr[f] * d2 + bias[f], 0.0f);
    acc += h * Wl[f];
  }
  out[i] = acc;
}

// ---------------------------------------------------------------- launch ----
extern "C" void kernel_launch(void* const* d_in, const int* in_sizes, int n_in,
                              void* d_out, int out_size, void* d_ws,
                              size_t ws_size, hipStream_t stream) {
  const float* x  = (const float*)d_in[0];
  const int*   ei = (const int*)d_in[1];   // JAX x64-off: int64 -> int32
  const float* W1 = (const float*)d_in[2];
  const float* b1 = (const float*)d_in[3];
  const float* W2 = (const float*)d_in[4];
  const float* b2 = (const float*)d_in[5];
  const float* Wl = (const float*)d_in[6];
  const float* bl = (const float*)d_in[7];
  float* out = (float*)d_out;

  int n = in_sizes[0] / FDIM;  // 1,000,000 nodes
  int e = in_sizes[1] / 2;     // 32,000,000 edges
  const int* src = ei;
  const int* dst = ei + e;

  // Workspace layout (all 256B aligned): dinv | bufA (xw/h) | bufB (accum)
  char*  ws   = (char*)d_ws;
  size_t off0 = 0;
  size_t off1 = (off0 + (size_t)n * 4 + 255) & ~(size_t)255;
  size_t off2 = (off1 + (size_t)n * PSTR * 4 + 255) & ~(size_t)255;
  float* dinv = (float*)(ws + off0);
  float* bufA = (float*)(ws + off1);
  float* bufB = (float*)(ws + off2);

  const int B  = 256;
  int gn = (n + B - 1) / B;
  int ge = (e + B - 1) / B;
  int gw = ((n + 15) / 16 * 32 + B - 1) / B;  // one wave per 16 nodes

  // degree (with self loops) -> deg_inv_sqrt; zero accumulators
  deg_init_kernel<<<gn, B, 0, stream>>>(dinv, bufB, n);
  deg_edge_kernel<<<ge, B, 0, stream>>>(dst, dinv, e);
  deg_fin_kernel<<<gn, B, 0, stream>>>(dinv, n);

  // layer 1
  wmma_transform_kernel<<<gw, B, 0, stream>>>(x, FDIM, W1, bufA, n);
  edge_agg_kernel<<<ge, B, 0, stream>>>(src, dst, dinv, bufA, bufB, e);
  finalize1_kernel<<<gn, B, 0, stream>>>(bufB, bufA, dinv, b1, n);

  // layer 2 (transform in place on bufA)
  wmma_transform_kernel<<<gw, B, 0, stream>>>(bufA, PSTR, W2, bufA, n);
  edge_agg_kernel<<<ge, B, 0, stream>>>(src, dst, dinv, bufA, bufB, e);

  // head
  finalize2_kernel<<<gn, B, 0, stream>>>(bufB, bufA, dinv, b2, Wl, bl, out, n);
}